// Detect_28552942584115
// MI455X (gfx1250) — compile-verified
//
#include <hip/hip_runtime.h>
#include <hip/hip_bf16.h>

// YOLO detect-head decode for MI455X (gfx1250):
//  - per level: 1x1-conv GEMM (M=267, K=Cin, N=B*HW) via V_WMMA_F32_16X16X4_F32
//  - activation panel staged to LDS by the Tensor Data Mover (tensor_load_to_lds),
//    double-buffered with s_wait_tensorcnt
//  - fused sigmoid/exp decode epilogue, scattered store into (b, 25200, 89)

typedef float v2f  __attribute__((ext_vector_type(2)));
typedef float v8f  __attribute__((ext_vector_type(8)));
typedef unsigned int u32x4 __attribute__((ext_vector_type(4)));
typedef int   i32x4 __attribute__((ext_vector_type(4)));
typedef int   i32x8 __attribute__((ext_vector_type(8)));

#define NO_  89
#define NA_  3
#define MROWS (NO_ * NA_)      // 267
#define TOTAL_ROWS 25200

// ---------------------------------------------------------------------------
// TDM: 2-D tile load Global -> LDS. Descriptor packed per CDNA5 ISA D# spec:
//  group0: [1:0]=count=1, [63:32]=lds_addr, [120:64]=global_addr, [127:126]=2
//  group1: [17:16]=data_size(2 -> 4B), [79:48]=tensor_dim0, [111:80]=tensor_dim1,
//          [127:112]=tile_dim0, [143:128]=tile_dim1, [207:160]=tensor_dim0_stride
// amdgpu-toolchain (clang-23) builtin arity: 6 args
//  (u32x4 g0, i32x8 g1, i32x4 g2, i32x4 g3, i32x8 extra, i32 cpol)
// ---------------------------------------------------------------------------
__device__ __forceinline__ void tdm_load_tile_2d(unsigned lds_byte_addr,
                                                 const float* gptr,
                                                 unsigned tile_x_elems,
                                                 unsigned tile_y_rows,
                                                 unsigned tensor_x_elems,
                                                 unsigned tensor_y_rows,
                                                 unsigned long long row_stride_elems)
{
    unsigned long long ga = (unsigned long long)(const void*)gptr;
    u32x4 g0;
    g0[0] = 1u;                                       // count=1, user descriptor
    g0[1] = lds_byte_addr;                            // lds_addr (bytes)
    g0[2] = (unsigned)(ga & 0xFFFFFFFFull);           // global_addr[31:0]
    g0[3] = ((unsigned)((ga >> 32) & 0x01FFFFFFull))  // global_addr[56:32]
          | 0x80000000u;                              // type = 2 ("image")
    i32x8 g1;
    g1[0] = (int)(2u << 16);                          // data_size = 2 -> 4 bytes
    g1[1] = (int)((tensor_x_elems & 0xFFFFu) << 16);                  // dim0[15:0]
    g1[2] = (int)(((tensor_x_elems >> 16) & 0xFFFFu)
               | ((tensor_y_rows & 0xFFFFu) << 16));                  // dim0[31:16] | dim1[15:0]
    g1[3] = (int)(((tensor_y_rows >> 16) & 0xFFFFu)
               | ((tile_x_elems & 0xFFFFu) << 16));                   // dim1[31:16] | tile_dim0
    g1[4] = (int)(tile_y_rows & 0xFFFFu);                             // tile_dim1, tile_dim2=0
    g1[5] = (int)(row_stride_elems & 0xFFFFFFFFull);                  // dim0_stride[31:0]
    g1[6] = (int)((row_stride_elems >> 32) & 0xFFFFull);              // dim0_stride[47:32]
    g1[7] = 0;
    i32x4 g2 = {0, 0, 0, 0};
    i32x4 g3 = {0, 0, 0, 0};
    i32x8 g4 = {0, 0, 0, 0, 0, 0, 0, 0};              // extra operand (zero-filled)
    __builtin_amdgcn_tensor_load_to_lds(g0, g1, g2, g3, g4, 0);
}

// ---------------------------------------------------------------------------
// One kernel per pyramid level.
//   block = 17 waves (544 threads); wave w owns M-tile w (rows 16w..16w+15)
//   block covers BN = 16*NT N-columns of one batch image
//   K loop in chunks of 32, TDM double-buffered in LDS
// ---------------------------------------------------------------------------
template <int NT>
__global__ __launch_bounds__(544)
void detect_head_kernel(const float* __restrict__ X, const float* __restrict__ Wt,
                        const float* __restrict__ Bs, float* __restrict__ Out,
                        int Cin, int HW, int nx, float strideF, int levelOff,
                        float a0x, float a0y, float a1x, float a1y, float a2x, float a2y)
{
    constexpr int BN = 16 * NT;     // N columns per block
    constexpr int KC = 32;          // K chunk
    __shared__ float ldsB[2][KC * BN];

    const int tid  = threadIdx.x;
    const int wv   = tid >> 5;      // 0..16 : M-tile index
    const int lane = tid & 31;
    const int h    = lane >> 4;     // lane half (per WMMA f32 layouts)
    const int nl   = lane & 15;

    const long long nBase = (long long)blockIdx.x * BN;
    const int bIdx = (int)(nBase / HW);                 // batch image (block never straddles)
    const int hw0  = (int)(nBase - (long long)bIdx * HW);

    // A-fragment row for this lane (16x4 f32 A layout: lane = M, vgpr+half = K)
    const int oRow   = wv * 16 + nl;
    const int oClamp = oRow < MROWS ? oRow : (MROWS - 1);
    const float* wRow = Wt + (size_t)oClamp * Cin + 2 * h;

    const float* gB = X + (size_t)bIdx * Cin * (size_t)HW + hw0;
    const unsigned ldsAddr0 = (unsigned)(unsigned long long)(const void*)&ldsB[0][0];
    const unsigned ldsAddr1 = (unsigned)(unsigned long long)(const void*)&ldsB[1][0];

    v8f acc[NT];
#pragma unroll
    for (int t = 0; t < NT; ++t)
#pragma unroll
        for (int r = 0; r < 8; ++r) acc[t][r] = 0.0f;

    const int nChunks = Cin / KC;

    // prologue: DMA chunk 0
    if (wv == 0)
        tdm_load_tile_2d(ldsAddr0, gB, BN, KC, (unsigned)HW, KC, (unsigned long long)HW);

    for (int c = 0; c < nChunks; ++c) {
        if (wv == 0) __builtin_amdgcn_s_wait_tensorcnt((short)0);   // chunk c landed
        __syncthreads();                                            // LDS visible to all waves
        if (wv == 0 && (c + 1) < nChunks)                           // prefetch chunk c+1
            tdm_load_tile_2d(((c + 1) & 1) ? ldsAddr1 : ldsAddr0,
                             gB + (size_t)(c + 1) * KC * HW,
                             BN, KC, (unsigned)HW, KC, (unsigned long long)HW);

        const float* bufB = &ldsB[c & 1][0];
        const float* wk   = wRow + c * KC;
#pragma unroll
        for (int s = 0; s < KC / 4; ++s) {
            // A: w[o, c*KC + 4s + 2h (+1)]   (even offset -> aligned b64 load)
            const v2f av = *(const v2f*)(wk + 4 * s);
            const int krow = 4 * s + 2 * h;
#pragma unroll
            for (int t = 0; t < NT; ++t) {
                v2f bv;
                bv.x = bufB[(krow    ) * BN + t * 16 + nl];
                bv.y = bufB[(krow + 1) * BN + t * 16 + nl];
                acc[t] = __builtin_amdgcn_wmma_f32_16x16x4_f32(
                    false, av, false, bv, (short)0, acc[t], false, false);
            }
        }
    }

    // ---------------- fused decode epilogue ----------------
    const float anch[6] = {a0x, a0y, a1x, a1y, a2x, a2y};
    const size_t rowB = ((size_t)bIdx * TOTAL_ROWS + levelOff) * NO_;
#pragma unroll
    for (int t = 0; t < NT; ++t) {
        const int hw = hw0 + t * 16 + nl;
        const float gx = (float)(hw % nx);
        const float gy = (float)(hw / nx);
#pragma unroll
        for (int r = 0; r < 8; ++r) {
            const int o = wv * 16 + 8 * h + r;      // C/D layout: vgpr r, half h, col nl
            if (o < MROWS) {
                const int a = o / NO_;
                const int j = o - a * NO_;
                const float v = acc[t][r] + Bs[o];
                float res;
                if (j >= 4) {
                    res = 1.0f / (1.0f + __expf(-v));                       // class/obj sigmoid
                } else if (j < 2) {
                    const float g = (j == 0) ? gx : gy;
                    res = (1.0f / (1.0f + __expf(-v)) + g) * strideF;       // xy
                } else {
                    res = __expf(v) * anch[a * 2 + (j - 2)];                // wh
                }
                Out[rowB + ((size_t)a * HW + hw) * NO_ + j] = res;
            }
        }
    }
}

// ---------------------------------------------------------------------------
extern "C" void kernel_launch(void* const* d_in, const int* in_sizes, int n_in,
                              void* d_out, int out_size, void* d_ws, size_t ws_size,
                              hipStream_t stream)
{
    (void)in_sizes; (void)n_in; (void)out_size; (void)d_ws; (void)ws_size;
    const float* x0 = (const float*)d_in[0];
    const float* w0 = (const float*)d_in[1];
    const float* b0 = (const float*)d_in[2];
    const float* x1 = (const float*)d_in[3];
    const float* w1 = (const float*)d_in[4];
    const float* b1 = (const float*)d_in[5];
    const float* x2 = (const float*)d_in[6];
    const float* w2 = (const float*)d_in[7];
    const float* b2 = (const float*)d_in[8];
    float* out = (float*)d_out;

    dim3 blk(544);   // 17 waves: one per 16-row M-tile of the 267-row output
    // L0: N = 8*6400 = 51200, 64 cols/block -> 800 blocks
    detect_head_kernel<4><<<dim3(800), blk, 0, stream>>>(
        x0, w0, b0, out, 256, 6400, 80, 8.0f, 0,
        10.f, 13.f, 16.f, 30.f, 33.f, 23.f);
    // L1: N = 8*1600 = 12800 -> 200 blocks
    detect_head_kernel<4><<<dim3(200), blk, 0, stream>>>(
        x1, w1, b1, out, 512, 1600, 40, 16.0f, 19200,
        30.f, 61.f, 62.f, 45.f, 59.f, 119.f);
    // L2: HW=400 not divisible by 64 -> 16 cols/block, 3200/16 = 200 blocks
    detect_head_kernel<1><<<dim3(200), blk, 0, stream>>>(
        x2, w2, b2, out, 1024, 400, 20, 32.0f, 24000,
        116.f, 90.f, 156.f, 198.f, 373.f, 326.f);
}